// MutualProjection_50689204027916
// MI455X (gfx1250) — compile-verified
//
#include <hip/hip_runtime.h>

typedef __attribute__((ext_vector_type(2))) float v2f;
typedef __attribute__((ext_vector_type(8))) float v8f;

#define IMG   128
#define BIGF  1000000.0f
#define NB    8
#define NV    4
#define NJ    21

// ---------------------------------------------------------------------------
// Kernel 1: one wave32 per batch b.
//   A(16x4): A[m=j*4+x, y] = inv[b,j,x,y]
//   B(4x16): B[y, n=i*4+z] = cam[b,i,y,z]
//   D = A*B  =>  D[j*4+x, i*4+z] = mutual[b,i,j,x,z]   (16 pose products / wmma)
// A layout (ISA 16x4 f32): lanes 0-15 hold K=0,1 in v0,v1; lanes 16-31 K=2,3.
// B layout (row striped across lanes per VGPR): VGPR p = rows {p, p+2},
//   lanes 0-15 -> N, lanes 16-31 -> N (second row half).
// D layout (16x16 f32 C/D): VGPR p, lanes 0-15 -> M=p, lanes 16-31 -> M=p+8.
// ---------------------------------------------------------------------------
__global__ __launch_bounds__(32)
void mutual_proj_wmma_kernel(const float* __restrict__ cam,     // [B,V,4,4]
                             const float* __restrict__ inv,     // [B,V,4,4]
                             const float* __restrict__ joints,  // [B,V,J,3]
                             float* __restrict__ projOut)       // [B,V,V,J,3]
{
    __shared__ float mut[16 * 16];   // mut[M*16+N] = D[M][N]

    const int b     = blockIdx.x;
    const int lane  = threadIdx.x;          // 0..31, EXEC all ones
    const int m     = lane & 15;            // row (A) / col (B) index
    const int halfK = (lane >> 4) << 1;     // 0 or 2

    // ---- A fragment: inv[b, j=m>>2, x=m&3, y=halfK..halfK+1]
    const float* invB = inv + b * 64;
    v2f afrag;
    afrag.x = invB[m * 4 + halfK];
    afrag.y = invB[m * 4 + halfK + 1];

    // ---- B fragment: cam[b, i=m>>2, y, z=m&3]
    const float* camB = cam + b * 64;
    const int iB = m >> 2, zB = m & 3;
    v2f bfrag;
    bfrag.x = camB[(iB * 4 + halfK) * 4 + zB];
    bfrag.y = camB[(iB * 4 + halfK + 1) * 4 + zB];

    v8f acc = {};
    acc = __builtin_amdgcn_wmma_f32_16x16x4_f32(
        /*neg_a=*/false, afrag, /*neg_b=*/false, bfrag,
        /*c_mod=*/(short)0, acc, /*reuse_a=*/false, /*reuse_b=*/false);

    // ---- scatter D to LDS
    const int mBase = (lane >> 4) * 8;
    #pragma unroll
    for (int p = 0; p < 8; ++p)
        mut[(mBase + p) * 16 + m] = acc[p];
    __syncthreads();

    // ---- project the 21 joints through all 16 (i,j) pose pairs
    // proj[b,i,j,k,x] = sum_y mut[(j*4+x)*16 + i*4+y] * joint_h[y], joint_h=(jx,jy,jz,1)
    for (int q = lane; q < NV * NV * NJ; q += 32) {
        const int i   = q / (NV * NJ);
        const int rem = q - i * (NV * NJ);
        const int j   = rem / NJ;
        const int k   = rem - j * NJ;

        const float* jp = joints + ((b * NV + i) * NJ + k) * 3;
        const float jx = jp[0], jy = jp[1], jz = jp[2];

        float* op = projOut + (((b * NV + i) * NV + j) * NJ + k) * 3;
        #pragma unroll
        for (int x = 0; x < 3; ++x) {
            const float* row = &mut[(j * 4 + x) * 16 + i * 4];
            op[x] = fmaf(row[0], jx, fmaf(row[1], jy, fmaf(row[2], jz, row[3])));
        }
    }
}

// ---------------------------------------------------------------------------
// Kernel 2: ball-depth rasterization + min over J.
// grid.x = B*V*V images, grid.y = 4 pixel slices (4096 px each), 256 threads.
// ---------------------------------------------------------------------------
__global__ __launch_bounds__(256)
void render_min_kernel(const float* __restrict__ proj,    // [B,V,V,J,3]
                       const float* __restrict__ radius,  // [J]
                       float* __restrict__ depth)         // [B,V,V,IMG,IMG]
{
    __shared__ float sx[NJ], sy[NJ], sz[NJ], sr2[NJ];

    const int img = blockIdx.x;
    const int tid = threadIdx.x;

    if (tid < NJ) {
        const float* p = proj + (img * NJ + tid) * 3;
        sx[tid] = p[0];
        sy[tid] = p[1];
        sz[tid] = p[2];
        const float r = radius[tid];
        sr2[tid] = r * r;
    }
    __syncthreads();

    float* out = depth + (size_t)img * (IMG * IMG);
    const int base = blockIdx.y * (IMG * IMG / 4);   // 4096-px slice

    #pragma unroll 4
    for (int it = 0; it < (IMG * IMG / 4) / 256; ++it) {
        const int p = base + it * 256 + tid;
        const float u = (float)(p & (IMG - 1));
        const float v = (float)((p >> 7) & (IMG - 1));

        float best = BIGF;
        #pragma unroll
        for (int k = 0; k < NJ; ++k) {
            const float dx = u - sx[k];
            const float dy = v - sy[k];
            const float d2 = fmaf(dx, dx, dy * dy);
            const float r2 = sr2[k];
            // mirrors reference: safe = where(inside, r2-d2, 1.0)
            const float safe = (d2 < r2) ? (r2 - d2) : 1.0f;
            const float dep  = (d2 < r2) ? (sz[k] - sqrtf(safe)) : BIGF;
            best = fminf(best, dep);
        }
        out[p] = best;
    }
}

// ---------------------------------------------------------------------------
extern "C" void kernel_launch(void* const* d_in, const int* in_sizes, int n_in,
                              void* d_out, int out_size, void* d_ws, size_t ws_size,
                              hipStream_t stream)
{
    (void)in_sizes; (void)n_in; (void)out_size; (void)d_ws; (void)ws_size;

    const float* cam    = (const float*)d_in[0];  // camera_poses     [8,4,4,4]
    const float* inv    = (const float*)d_in[1];  // inv_camera_poses [8,4,4,4]
    const float* joints = (const float*)d_in[2];  // joints           [8,4,21,3]
    const float* rad    = (const float*)d_in[3];  // radiuses         [21]

    float* depth   = (float*)d_out;                                   // 2,097,152 f32
    float* projOut = depth + (size_t)NB * NV * NV * IMG * IMG;        // + 8,064 f32

    mutual_proj_wmma_kernel<<<dim3(NB), dim3(32), 0, stream>>>(cam, inv, joints, projOut);

    render_min_kernel<<<dim3(NB * NV * NV, 4), dim3(256), 0, stream>>>(projOut, rad, depth);
}